// EdgeConvUp_67997922230595
// MI455X (gfx1250) — compile-verified
//
#include <hip/hip_runtime.h>
#include <math.h>

// ---- problem constants (match reference) ----
#define N_REF  50000
#define N_Q    25000
#define E_NUM  400000
#define C0     128
#define C1     256
#define C2     256
#define BN_EPS 1e-5f

typedef __attribute__((ext_vector_type(2))) float v2f;
typedef __attribute__((ext_vector_type(8))) float v8f;
typedef int v4i_b __attribute__((vector_size(16)));   // matches builtin's pointee type

// ---- CDNA5 async global->LDS copy (ASYNCcnt path), with safe fallback ----
#if __has_builtin(__builtin_amdgcn_global_load_async_to_lds_b128) && \
    __has_builtin(__builtin_amdgcn_s_wait_asynccnt)
#define HAVE_ASYNC_LDS 1
typedef __attribute__((address_space(1))) v4i_b* async_gptr_t;
typedef __attribute__((address_space(3))) v4i_b* async_lptr_t;
#endif

__device__ __forceinline__ void lds_copy16(const float* g, float* l) {
#ifdef HAVE_ASYNC_LDS
  __builtin_amdgcn_global_load_async_to_lds_b128(
      (async_gptr_t)(void*)g, (async_lptr_t)l, 0, 0);
#else
  *(float4*)l = *(const float4*)g;
#endif
}

__device__ __forceinline__ void lds_copy_wait() {
#ifdef HAVE_ASYNC_LDS
  __builtin_amdgcn_s_wait_asynccnt(0);   // s_wait_asynccnt 0x0
#endif
}

// ---------------------------------------------------------------------------
// GEMM C[M,N] = A[M,K] @ B[K,N], f32, via V_WMMA_F32_16X16X4_F32.
// Block = 128 threads = 4 waves. Block tile = 64 rows x 64 cols:
//   wave w owns rows blockIdx.x*64 + w*16, all 64 cols -> 4 accumulators,
//   one A fragment per K-step feeds all 4 WMMAs.
// B tile staged to LDS in 128-row chunks via async-to-LDS (row stride padded
// to 72 floats: half-waves read rows k / k+2 -> disjoint bank groups).
// Fragment layouts per CDNA5 ISA 7.12.2:
//   A 16x4 f32 : lane&15 = M, vgpr j holds K = 2*(lane>>4)+j
//   B 4x16 f32 : lane&15 = N, vgpr j holds K = 2*(lane>>4)+j
//   C/D 16x16  : lane&15 = N, vgpr i holds M = i + 8*(lane>>4)
// ---------------------------------------------------------------------------
#define KSTAGE  128
#define BSTRIDE 72

__global__ __launch_bounds__(128)
void gemm_wmma_f32(const float* __restrict__ A, const float* __restrict__ B,
                   float* __restrict__ C, int M, int K, int N) {
  __shared__ float Bs[KSTAGE * BSTRIDE];          // 36 KB

  const int lane  = threadIdx.x & 31;
  const int wave  = threadIdx.x >> 5;
  const int hi    = lane >> 4;                    // 0 or 1
  const int khalf = hi * 2;                       // 0 or 2
  const int j     = lane & 15;
  const int m0w   = blockIdx.x * 64 + wave * 16;  // this wave's row tile
  const int n0    = blockIdx.y * 64;              // block's column tile

  int mrow  = m0w + j;
  int mload = mrow < M ? mrow : (M - 1);          // clamp: EXEC stays all-1s
  const float* __restrict__ Arow = A + (size_t)mload * K;

  // cooperative staging indices: 8 rows per pass, 16 float4 chunks per row
  const int srow = threadIdx.x >> 4;              // 0..7
  const int sj4  = (threadIdx.x & 15) * 4;        // 0..60

  v8f acc0 = {}, acc1 = {}, acc2 = {}, acc3 = {};

  for (int ks = 0; ks < K; ks += KSTAGE) {        // K is a multiple of 128 here
    // ---- stage B[ks..ks+127][n0..n0+63] -> Bs (async global->LDS) ----
#pragma unroll 4
    for (int it = 0; it < KSTAGE / 8; ++it) {
      int row = it * 8 + srow;
      lds_copy16(B + (size_t)(ks + row) * N + n0 + sj4,
                 &Bs[row * BSTRIDE + sj4]);
    }
    lds_copy_wait();
    __syncthreads();

    // ---- compute: 32 K-steps, 4 WMMAs each, single A fragment reused ----
    for (int k = 0; k < KSTAGE; k += 4) {
      __builtin_prefetch(Arow + ks + k + 16, 0, 0);     // global_prefetch_b8
      v2f a = *(const v2f*)(Arow + ks + k + khalf);     // 8B-aligned
      const float* brow = &Bs[(k + khalf) * BSTRIDE + j];
      v2f b0; b0.x = brow[0];   b0.y = brow[BSTRIDE];
      v2f b1; b1.x = brow[16];  b1.y = brow[BSTRIDE + 16];
      v2f b2; b2.x = brow[32];  b2.y = brow[BSTRIDE + 32];
      v2f b3; b3.x = brow[48];  b3.y = brow[BSTRIDE + 48];
      acc0 = __builtin_amdgcn_wmma_f32_16x16x4_f32(false, a, false, b0, (short)0, acc0, false, false);
      acc1 = __builtin_amdgcn_wmma_f32_16x16x4_f32(false, a, false, b1, (short)0, acc1, false, false);
      acc2 = __builtin_amdgcn_wmma_f32_16x16x4_f32(false, a, false, b2, (short)0, acc2, false, false);
      acc3 = __builtin_amdgcn_wmma_f32_16x16x4_f32(false, a, false, b3, (short)0, acc3, false, false);
    }
    __syncthreads();                              // before re-staging Bs
  }

  // ---- store: D vgpr i -> row m0w + i + 8*hi ----
  const int rbase = m0w + hi * 8;
#pragma unroll
  for (int i = 0; i < 8; ++i) {
    int r = rbase + i;
    if (r < M) {
      float* crow = C + (size_t)r * N + n0 + j;
      crow[0]  = acc0[i];
      crow[16] = acc1[i];
      crow[32] = acc2[i];
      crow[48] = acc3[i];
    }
  }
}

// ---------------------------------------------------------------------------
// Per-edge inverse-distance weights + segment-sum of weights per query node.
// ---------------------------------------------------------------------------
__global__ void edge_weights(const float* __restrict__ rb, const float* __restrict__ qb,
                             const int* __restrict__ er, const int* __restrict__ eq,
                             float* __restrict__ w, float* __restrict__ wsum) {
  int e = blockIdx.x * blockDim.x + threadIdx.x;
  if (e >= E_NUM) return;
  int r = er[e], q = eq[e];
  float4 rv = ((const float4*)rb)[r];   // bxyz rows are 16B aligned
  float4 qv = ((const float4*)qb)[q];
  float dx = rv.y - qv.y, dy = rv.z - qv.z, dz = rv.w - qv.w;
  float d  = sqrtf(dx * dx + dy * dy + dz * dz);
  float wv = 1.0f / (d + 1e-8f);
  w[e] = wv;
  atomicAdd(&wsum[q], wv);
}

// ---------------------------------------------------------------------------
// Column (per-channel) sum / sum-of-squares for training-mode BN statistics.
// blockDim.x == C (128 or 256); each block reduces 64 rows then atomics.
// ---------------------------------------------------------------------------
__global__ void col_stats(const float* __restrict__ X, int M, int C,
                          float* __restrict__ sum, float* __restrict__ sumsq) {
  int c  = threadIdx.x;
  int r0 = blockIdx.x * 64;
  int r1 = (r0 + 64 < M) ? (r0 + 64) : M;
  float s = 0.f, s2 = 0.f;
  for (int r = r0; r < r1; ++r) {
    float v = X[(size_t)r * C + c];
    s += v; s2 += v * v;
  }
  atomicAdd(&sum[c], s);
  atomicAdd(&sumsq[c], s2);
}

// Fold BN into per-channel affine: y = x*a + b, a = g*rsqrt(var+eps),
// b = beta - mean*a.  (Pre-BN channel biases cancel -> b1/b2 dropped.)
__global__ void bn_finalize(const float* __restrict__ sum, const float* __restrict__ sumsq,
                            const float* __restrict__ g, const float* __restrict__ beta,
                            float invM, int C, float* __restrict__ a, float* __restrict__ b) {
  int c = blockIdx.x * blockDim.x + threadIdx.x;
  if (c >= C) return;
  float m   = sum[c] * invM;
  float var = sumsq[c] * invM - m * m;
  float s   = rsqrtf(var + BN_EPS);
  float ac  = g[c] * s;
  a[c] = ac;
  b[c] = beta[c] - m * ac;
}

// ---------------------------------------------------------------------------
// Gather-weight-scatter: query_f[q] += BN(tmp1[r]) * (w_e / wsum[q]).
// One wave per edge (32 lanes x 4 channels). Edge index is wave-uniform ->
// readfirstlane forces the index/weight loads onto the scalar path.
// ---------------------------------------------------------------------------
__global__ void edge_scatter(const float* __restrict__ tmp1,
                             const float* __restrict__ a0, const float* __restrict__ b0,
                             const int* __restrict__ er, const int* __restrict__ eq,
                             const float* __restrict__ w, const float* __restrict__ wsum,
                             float* __restrict__ qf) {
  unsigned t = blockIdx.x * blockDim.x + threadIdx.x;
  if (t >= (unsigned)E_NUM * 32u) return;
  int e = __builtin_amdgcn_readfirstlane((int)(t >> 5));   // wave-uniform
  int c = (t & 31) * 4;
  int r = er[e], q = eq[e];
  float wn = w[e] / wsum[q];
  float4 v = *(const float4*)(tmp1 + (size_t)r * C0 + c);
  float* dst = qf + (size_t)q * C0 + c;
  atomicAdd(dst + 0, (v.x * a0[c + 0] + b0[c + 0]) * wn);
  atomicAdd(dst + 1, (v.y * a0[c + 1] + b0[c + 1]) * wn);
  atomicAdd(dst + 2, (v.z * a0[c + 2] + b0[c + 2]) * wn);
  atomicAdd(dst + 3, (v.w * a0[c + 3] + b0[c + 3]) * wn);
}

// x0 = relu(query_f + BN(tmp2))  (skip branch join), float4-vectorized
__global__ void make_x0(const float* __restrict__ qf, const float* __restrict__ tmp2,
                        const float* __restrict__ as_, const float* __restrict__ bs_,
                        float* __restrict__ x0) {
  int i4 = blockIdx.x * blockDim.x + threadIdx.x;
  if (i4 >= (N_Q * C0) / 4) return;
  int c = (i4 * 4) & (C0 - 1);
  float4 qv = ((const float4*)qf)[i4];
  float4 tv = ((const float4*)tmp2)[i4];
  float4 o;
  o.x = fmaxf(qv.x + tv.x * as_[c + 0] + bs_[c + 0], 0.f);
  o.y = fmaxf(qv.y + tv.y * as_[c + 1] + bs_[c + 1], 0.f);
  o.z = fmaxf(qv.z + tv.z * as_[c + 2] + bs_[c + 2], 0.f);
  o.w = fmaxf(qv.w + tv.w * as_[c + 3] + bs_[c + 3], 0.f);
  ((float4*)x0)[i4] = o;
}

// y = relu(x*a + b), C power of two (cmask = C-1), float4-vectorized
__global__ void bn_relu(const float* __restrict__ X, const float* __restrict__ a,
                        const float* __restrict__ b, float* __restrict__ Y,
                        int n4, int cmask) {
  int i4 = blockIdx.x * blockDim.x + threadIdx.x;
  if (i4 >= n4) return;
  int c = (i4 * 4) & cmask;
  float4 xv = ((const float4*)X)[i4];
  float4 o;
  o.x = fmaxf(xv.x * a[c + 0] + b[c + 0], 0.f);
  o.y = fmaxf(xv.y * a[c + 1] + b[c + 1], 0.f);
  o.z = fmaxf(xv.z * a[c + 2] + b[c + 2], 0.f);
  o.w = fmaxf(xv.w * a[c + 3] + b[c + 3], 0.f);
  ((float4*)Y)[i4] = o;
}

// ---------------------------------------------------------------------------
extern "C" void kernel_launch(void* const* d_in, const int* in_sizes, int n_in,
                              void* d_out, int out_size, void* d_ws, size_t ws_size,
                              hipStream_t stream) {
  const float* ref_bxyz   = (const float*)d_in[0];
  const float* ref_feat   = (const float*)d_in[1];
  const float* query_bxyz = (const float*)d_in[2];
  const float* query_feat = (const float*)d_in[3];
  const int*   e_ref      = (const int*)d_in[4];
  const int*   e_query    = (const int*)d_in[5];
  const float* Wf0 = (const float*)d_in[6];
  const float* gf0 = (const float*)d_in[7];
  const float* bf0 = (const float*)d_in[8];
  const float* Ws0 = (const float*)d_in[9];
  const float* gs0 = (const float*)d_in[10];
  const float* bs0 = (const float*)d_in[11];
  const float* W1    = (const float*)d_in[12];
  // d_in[13] = b1: cancelled by training-mode BN mean subtraction
  const float* g1    = (const float*)d_in[14];
  const float* beta1 = (const float*)d_in[15];
  const float* W2    = (const float*)d_in[16];
  // d_in[17] = b2: cancelled likewise
  const float* g2    = (const float*)d_in[18];
  const float* beta2 = (const float*)d_in[19];
  float* out = (float*)d_out;

  // ---- workspace layout (floats). Front section is zero-initialized. ----
  float* ws = (float*)d_ws;
  size_t o = 0;
  float* w_sum = ws + o; o += N_Q;
  float* sum0 = ws + o; o += C0;            // BN stats accumulators
  float* sq0  = ws + o; o += C0;
  float* sumS = ws + o; o += C0;
  float* sqS  = ws + o; o += C0;
  float* sum1 = ws + o; o += C1;
  float* sq1  = ws + o; o += C1;
  float* sum2 = ws + o; o += C2;
  float* sq2  = ws + o; o += C2;
  float* a0 = ws + o; o += C0;  float* b0 = ws + o; o += C0;   // folded BN affines
  float* aS = ws + o; o += C0;  float* bS = ws + o; o += C0;
  float* a1 = ws + o; o += C1;  float* b1f = ws + o; o += C1;
  float* a2 = ws + o; o += C2;  float* b2f = ws + o; o += C2;
  float* query_f = ws + o; o += (size_t)N_Q * C0;               // scatter accumulator
  const size_t zero_bytes = o * sizeof(float);                  // everything above -> 0
  float* wbuf = ws + o; o += E_NUM;
  float* tmp1 = ws + o; o += (size_t)N_REF * C0;
  float* tmp2 = ws + o; o += (size_t)N_Q * C0;
  float* x0   = ws + o; o += (size_t)N_Q * C0;
  float* h1   = ws + o; o += (size_t)N_Q * C1;
  float* h1r  = ws + o; o += (size_t)N_Q * C1;
  float* h2   = ws + o; o += (size_t)N_Q * C2;
  (void)in_sizes; (void)n_in; (void)out_size; (void)ws_size;

  (void)hipMemsetAsync(d_ws, 0, zero_bytes, stream);

  // 1) edge weights + per-query weight sums
  edge_weights<<<(E_NUM + 255) / 256, 256, 0, stream>>>(
      ref_bxyz, query_bxyz, e_ref, e_query, wbuf, w_sum);

  // 2) ref branch: tmp1 = ref_feat @ Wf0, BN stats over 50000 rows
  gemm_wmma_f32<<<dim3((N_REF + 63) / 64, C0 / 64), 128, 0, stream>>>(
      ref_feat, Wf0, tmp1, N_REF, C0, C0);
  col_stats<<<(N_REF + 63) / 64, C0, 0, stream>>>(tmp1, N_REF, C0, sum0, sq0);
  bn_finalize<<<1, C0, 0, stream>>>(sum0, sq0, gf0, bf0, 1.0f / N_REF, C0, a0, b0);

  // 3) gather-weight-scatter to query nodes (one wave per edge)
  edge_scatter<<<((unsigned)E_NUM * 32u + 255) / 256, 256, 0, stream>>>(
      tmp1, a0, b0, e_ref, e_query, wbuf, w_sum, query_f);

  // 4) skip branch: tmp2 = query_feat @ Ws0, BN stats over 25000 rows
  gemm_wmma_f32<<<dim3((N_Q + 63) / 64, C0 / 64), 128, 0, stream>>>(
      query_feat, Ws0, tmp2, N_Q, C0, C0);
  col_stats<<<(N_Q + 63) / 64, C0, 0, stream>>>(tmp2, N_Q, C0, sumS, sqS);
  bn_finalize<<<1, C0, 0, stream>>>(sumS, sqS, gs0, bs0, 1.0f / N_Q, C0, aS, bS);

  // 5) join + relu
  make_x0<<<((N_Q * C0 / 4) + 255) / 256, 256, 0, stream>>>(query_f, tmp2, aS, bS, x0);

  // 6) layer 1: h1 = x0 @ W1 (b1 cancels in BN), BN+relu
  gemm_wmma_f32<<<dim3((N_Q + 63) / 64, C1 / 64), 128, 0, stream>>>(
      x0, W1, h1, N_Q, C0, C1);
  col_stats<<<(N_Q + 63) / 64, C1, 0, stream>>>(h1, N_Q, C1, sum1, sq1);
  bn_finalize<<<1, C1, 0, stream>>>(sum1, sq1, g1, beta1, 1.0f / N_Q, C1, a1, b1f);
  bn_relu<<<((N_Q * C1 / 4) + 255) / 256, 256, 0, stream>>>(
      h1, a1, b1f, h1r, N_Q * C1 / 4, C1 - 1);

  // 7) layer 2: h2 = h1r @ W2 (b2 cancels in BN), BN+relu -> output
  gemm_wmma_f32<<<dim3((N_Q + 63) / 64, C2 / 64), 128, 0, stream>>>(
      h1r, W2, h2, N_Q, C1, C2);
  col_stats<<<(N_Q + 63) / 64, C2, 0, stream>>>(h2, N_Q, C2, sum2, sq2);
  bn_finalize<<<1, C2, 0, stream>>>(sum2, sq2, g2, beta2, 1.0f / N_Q, C2, a2, b2f);
  bn_relu<<<((N_Q * C2 / 4) + 255) / 256, 256, 0, stream>>>(
      h2, a2, b2f, out, N_Q * C2 / 4, C2 - 1);
}